// TypeNorm_51488067944936
// MI455X (gfx1250) — compile-verified
//
#include <hip/hip_runtime.h>
#include <stdint.h>

// TypeNorm: per-row LayerNorm (D=128) + per-type affine (T=8).
// Pure HBM-streaming kernel: ~2.06 GB traffic, ~88us floor at 23.3 TB/s.
// One wave32 per row; lane holds float4 -> one coalesced B128 per row.
// gamma/beta staged to LDS with CDNA5 async-DMA (global_load_async_to_lds_b128
// + s_wait_asynccnt); feature/output traffic uses non-temporal hints.

typedef float v4f __attribute__((ext_vector_type(4)));

#define DDIM 128
#define TMAX 8
#define WAVES_PER_BLOCK 8
#define BLOCK_THREADS 256
#define MAX_BLOCKS 8192

__global__ __launch_bounds__(BLOCK_THREADS) void typenorm_kernel(
    const int* __restrict__ type_list,
    const float* __restrict__ x,
    const float* __restrict__ gamma,
    const float* __restrict__ beta,
    float* __restrict__ out,
    int N,            // number of rows
    int n_par_vec4)   // T*D/4 float4s per param tensor (<= 256)
{
  __shared__ __align__(16) float s_gamma[TMAX * DDIM];
  __shared__ __align__(16) float s_beta [TMAX * DDIM];

  const int tid  = (int)threadIdx.x;
  const int wave = tid >> 5;   // wave32 on gfx1250
  const int lane = tid & 31;

  // ---- Stage per-type LayerNorm params into LDS via CDNA5 async DMA ----
  // 256 threads x one float4 each covers T*D = 1024 floats per tensor.
  if (tid < n_par_vec4) {
    uint64_t g_src = (uint64_t)(uintptr_t)gamma + (uint64_t)tid * 16u;
    uint64_t b_src = (uint64_t)(uintptr_t)beta  + (uint64_t)tid * 16u;
    uint32_t g_dst = (uint32_t)(uintptr_t)(&s_gamma[0]) + (uint32_t)tid * 16u;
    uint32_t b_dst = (uint32_t)(uintptr_t)(&s_beta[0])  + (uint32_t)tid * 16u;
    asm volatile("global_load_async_to_lds_b128 %0, %1, off"
                 :: "v"(g_dst), "v"(g_src) : "memory");
    asm volatile("global_load_async_to_lds_b128 %0, %1, off"
                 :: "v"(b_dst), "v"(b_src) : "memory");
  }
  // Drain this wave's ASYNCcnt, then barrier so every wave sees the LDS image.
  asm volatile("s_wait_asynccnt 0x0" ::: "memory");
  __syncthreads();

  const float inv_d = 1.0f / (float)DDIM;

  // Grid-stride over rows: one row per wave per iteration.
  const int row0   = (int)blockIdx.x * WAVES_PER_BLOCK + wave;
  const int rstep  = (int)gridDim.x * WAVES_PER_BLOCK;

  for (int row = row0; row < N; row += rstep) {
    // Coalesced, non-temporal 512B row read: 32 lanes x float4.
    const v4f* __restrict__ xrow = (const v4f*)(x + (size_t)row * DDIM);
    const v4f v = __builtin_nontemporal_load(xrow + lane);

    // Per-lane partial sums, then wave32 butterfly reduction.
    float s  = (v.x + v.y) + (v.z + v.w);
    float s2 = fmaf(v.x, v.x, fmaf(v.y, v.y, fmaf(v.z, v.z, v.w * v.w)));
#pragma unroll
    for (int m = 16; m >= 1; m >>= 1) {
      s  += __shfl_xor(s,  m, 32);
      s2 += __shfl_xor(s2, m, 32);
    }

    const float mean = s * inv_d;
    const float var  = fmaf(-mean, mean, s2 * inv_d);  // E[x^2] - mean^2
    const float rstd = rsqrtf(var + 1e-5f);

    // Per-type affine params from LDS (ds_load_b128 x2).
    const int t = type_list[row];
    const v4f g = *((const v4f*)s_gamma + (size_t)t * (DDIM / 4) + lane);
    const v4f b = *((const v4f*)s_beta  + (size_t)t * (DDIM / 4) + lane);

    v4f o;
    o.x = fmaf((v.x - mean) * rstd, g.x, b.x);
    o.y = fmaf((v.y - mean) * rstd, g.y, b.y);
    o.z = fmaf((v.z - mean) * rstd, g.z, b.z);
    o.w = fmaf((v.w - mean) * rstd, g.w, b.w);

    // Non-temporal coalesced 512B row write.
    __builtin_nontemporal_store(o, (v4f*)(out + (size_t)row * DDIM) + lane);
  }
}

extern "C" void kernel_launch(void* const* d_in, const int* in_sizes, int n_in,
                              void* d_out, int out_size, void* d_ws, size_t ws_size,
                              hipStream_t stream) {
  // setup_inputs() order: type_list, abstract_features, gamma, beta
  const int*   type_list = (const int*)  d_in[0];
  const float* x         = (const float*)d_in[1];
  const float* gamma     = (const float*)d_in[2];
  const float* beta      = (const float*)d_in[3];
  float*       out       = (float*)d_out;

  const int N = in_sizes[0];                 // rows
  int n_par_vec4 = in_sizes[2] / 4;          // T*D/4 float4s (== 256 for T=8,D=128)
  if (n_par_vec4 > BLOCK_THREADS) n_par_vec4 = BLOCK_THREADS;

  int rowgroups = (N + WAVES_PER_BLOCK - 1) / WAVES_PER_BLOCK;
  int blocks = rowgroups < MAX_BLOCKS ? rowgroups : MAX_BLOCKS;
  if (blocks < 1) blocks = 1;

  typenorm_kernel<<<dim3(blocks), dim3(BLOCK_THREADS), 0, stream>>>(
      type_list, x, gamma, beta, out, N, n_par_vec4);
}